// FreNonLocal_66305705115918
// MI455X (gfx1250) — compile-verified
//
#include <hip/hip_runtime.h>
#include <math.h>

typedef __attribute__((ext_vector_type(2))) float v2f;
typedef __attribute__((ext_vector_type(8))) float v8f;

#define PI_F 3.14159265358979323846f

__device__ __forceinline__ v8f wmma4(v2f a, v2f b, v8f c) {
    // D(16x16,f32) = A(16x4,f32) * B(4x16,f32) + C
    return __builtin_amdgcn_wmma_f32_16x16x4_f32(false, a, false, b, (short)0, c, false, false);
}

__device__ __forceinline__ v2f nt_load_v2f(const float* p) {
    return __builtin_nontemporal_load((const v2f*)p);
}
__device__ __forceinline__ float nt_load_f(const float* p) {
    return __builtin_nontemporal_load(p);
}
__device__ __forceinline__ void nt_store_f(float* p, float v) {
    __builtin_nontemporal_store(v, p);
}

// ---------------------------------------------------------------------------
// Kernel 0: build DCT basis Dh, Dv (each 128 x 16, row-major [h*16+u]).
// Dh[h,u] = (u==0) ? 1/sqrt(128) : sqrt(2/128) * cos(u*(2h+1)*pi/256)
// (reference's Dv uses sqrt(H) scale; H==W==128 so identical formula)
// ---------------------------------------------------------------------------
__global__ void build_basis_kernel(float* __restrict__ Dh, float* __restrict__ Dv) {
    for (int i = threadIdx.x; i < 128 * 16; i += blockDim.x) {
        int h = i >> 4, u = i & 15;
        float val;
        if (u == 0)
            val = 0.088388347648318447f;            // 1/sqrt(128)
        else
            val = 0.125f * __cosf((float)u * (2.0f * (float)h + 1.0f) * PI_F / 256.0f);
        Dh[i] = val;
        Dv[i] = val;
    }
}

// ---------------------------------------------------------------------------
// Kernel 1: forward separable DCT.  One block per (b,c) image (4096 blocks,
// 8 waves).  Wave w: T[16w:16w+16, :] = X[16w:16w+16, :] @ Dv  (K=128).
// Then wave 0: xf(16x16) = Dh^T @ T  (K=128).  xf flat index = u*16+v.
// x is a 256 MB single-use stream -> non-temporal loads.
// ---------------------------------------------------------------------------
__global__ void fwd_dct_kernel(const float* __restrict__ x,
                               const float* __restrict__ Dh,
                               const float* __restrict__ Dv,
                               float* __restrict__ xf) {
    __shared__ float Dvlds[128 * 16];
    __shared__ float Tlds[128 * 16];

    const int tid  = threadIdx.x;
    const int lane = tid & 31;
    const int wave = tid >> 5;
    const int half = lane >> 4;   // 0 | 1
    const int l15  = lane & 15;

    // stage Dv into LDS once per block
    for (int i = tid; i < 128 * 16; i += 256) Dvlds[i] = Dv[i];
    __syncthreads();

    const int bc = blockIdx.x;                       // b*512 + c
    const float* __restrict__ X = x + (size_t)bc * 16384;

    // ---- stage 1: T strip = X strip @ Dv ----
    v8f acc = {};
    const float* Xrow = X + (16 * wave + l15) * 128; // this lane's A row
    for (int k4 = 0; k4 < 128; k4 += 4) {
        v2f a = nt_load_v2f(Xrow + k4 + 2 * half);               // A[m][k]=X[row,k]
        v2f b;
        b.x = Dvlds[(k4 + 2 * half + 0) * 16 + l15];             // B[k][n]=Dv[k,n]
        b.y = Dvlds[(k4 + 2 * half + 1) * 16 + l15];
        acc = wmma4(a, b, acc);
    }
#pragma unroll
    for (int r = 0; r < 8; ++r)
        Tlds[(16 * wave + r + 8 * half) * 16 + l15] = acc[r];
    __syncthreads();

    // ---- stage 2: xf = Dh^T @ T ----
    if (wave == 0) {
        v8f acc2 = {};
        for (int k4 = 0; k4 < 128; k4 += 4) {
            v2f a;                                               // A[u][k]=Dh[k,u]
            a.x = Dh[(k4 + 2 * half + 0) * 16 + l15];
            a.y = Dh[(k4 + 2 * half + 1) * 16 + l15];
            v2f b;                                               // B[k][v]=T[k,v]
            b.x = Tlds[(k4 + 2 * half + 0) * 16 + l15];
            b.y = Tlds[(k4 + 2 * half + 1) * 16 + l15];
            acc2 = wmma4(a, b, acc2);
        }
        float* __restrict__ dst = xf + (size_t)bc * 256;
#pragma unroll
        for (int r = 0; r < 8; ++r)
            dst[(r + 8 * half) * 16 + l15] = acc2[r];
    }
}

// ---------------------------------------------------------------------------
// Kernel 2: q/k/v projection in DCT domain.
// q[b,o,l] = sum_c Wq[o,c]*xf[b,c,l] + bq[o]   (M=256,N=256,K=512) per batch.
// One 16x16 output tile per wave; B-fragment (xf) shared by 3 accumulators.
// 2048 waves = 256 blocks x 8 waves.  All operands are L2-resident.
// ---------------------------------------------------------------------------
__global__ void proj_qkv_kernel(const float* __restrict__ xf,
                                const float* __restrict__ Wq, const float* __restrict__ bq,
                                const float* __restrict__ Wk, const float* __restrict__ bk,
                                const float* __restrict__ Wv, const float* __restrict__ bv,
                                float* __restrict__ q, float* __restrict__ kk,
                                float* __restrict__ v) {
    const int lane = threadIdx.x & 31;
    const int wave = threadIdx.x >> 5;
    const int half = lane >> 4;
    const int l15  = lane & 15;

    const int id = blockIdx.x * 8 + wave;        // 0..2047
    const int b  = id >> 8;
    const int mt = (id >> 4) & 15;
    const int nt = id & 15;

    const float* __restrict__ xfB = xf + (size_t)b * 512 * 256;
    const int arow = (16 * mt + l15) * 512 + 2 * half;

    v8f aq = {}, ak = {}, av = {};
    for (int c4 = 0; c4 < 512; c4 += 4) {
        v2f bb;                                   // B[c][l]=xf[b,c,16nt+l]
        bb.x = xfB[(c4 + 2 * half + 0) * 256 + 16 * nt + l15];
        bb.y = xfB[(c4 + 2 * half + 1) * 256 + 16 * nt + l15];
        v2f a0 = *(const v2f*)(Wq + arow + c4);   // A[o][c]=W[o,c]
        v2f a1 = *(const v2f*)(Wk + arow + c4);
        v2f a2 = *(const v2f*)(Wv + arow + c4);
        aq = wmma4(a0, bb, aq);
        ak = wmma4(a1, bb, ak);
        av = wmma4(a2, bb, av);
    }
#pragma unroll
    for (int r = 0; r < 8; ++r) {
        const int m = 16 * mt + r + 8 * half;
        const size_t idx = ((size_t)b * 256 + m) * 256 + 16 * nt + l15;
        q[idx]  = aq[r] + bq[m];
        kk[idx] = ak[r] + bk[m];
        v[idx]  = av[r] + bv[m];
    }
}

// ---------------------------------------------------------------------------
// Kernel 3: fatt[b,k,l] = (1/HW) * sum_c kk[b,c,k] * q[b,c,l]
// M=256(k), N=256(l), K=256(c).  A is a transposed read of kk.
// ---------------------------------------------------------------------------
__global__ void fatt_kernel(const float* __restrict__ q,
                            const float* __restrict__ kk,
                            float* __restrict__ fatt) {
    const int lane = threadIdx.x & 31;
    const int wave = threadIdx.x >> 5;
    const int half = lane >> 4;
    const int l15  = lane & 15;

    const int id = blockIdx.x * 8 + wave;        // 0..2047
    const int b  = id >> 8;
    const int mt = (id >> 4) & 15;
    const int nt = id & 15;

    const size_t base = (size_t)b * 256 * 256;
    v8f acc = {};
    for (int c4 = 0; c4 < 256; c4 += 4) {
        v2f a;                                    // A[k][c]=kk[b,c,16mt+k]
        a.x = kk[base + (size_t)(c4 + 2 * half + 0) * 256 + 16 * mt + l15];
        a.y = kk[base + (size_t)(c4 + 2 * half + 1) * 256 + 16 * mt + l15];
        v2f bb;                                   // B[c][l]=q[b,c,16nt+l]
        bb.x = q[base + (size_t)(c4 + 2 * half + 0) * 256 + 16 * nt + l15];
        bb.y = q[base + (size_t)(c4 + 2 * half + 1) * 256 + 16 * nt + l15];
        acc = wmma4(a, bb, acc);
    }
    const float scale = 1.0f / 16384.0f;          // 1/HW
#pragma unroll
    for (int r = 0; r < 8; ++r) {
        const int m = 16 * mt + r + 8 * half;
        fatt[base + (size_t)m * 256 + 16 * nt + l15] = acc[r] * scale;
    }
}

// ---------------------------------------------------------------------------
// Kernel 4: y[b,c,l] = sum_k v[b,c,k] * fatt[b,k,l]   (M=256,N=256,K=256)
// ---------------------------------------------------------------------------
__global__ void attn_apply_kernel(const float* __restrict__ v,
                                  const float* __restrict__ fatt,
                                  float* __restrict__ y) {
    const int lane = threadIdx.x & 31;
    const int wave = threadIdx.x >> 5;
    const int half = lane >> 4;
    const int l15  = lane & 15;

    const int id = blockIdx.x * 8 + wave;
    const int b  = id >> 8;
    const int mt = (id >> 4) & 15;
    const int nt = id & 15;

    const size_t base = (size_t)b * 256 * 256;
    v8f acc = {};
    for (int k4 = 0; k4 < 256; k4 += 4) {
        v2f a = *(const v2f*)(v + base + (size_t)(16 * mt + l15) * 256 + k4 + 2 * half);
        v2f bb;
        bb.x = fatt[base + (size_t)(k4 + 2 * half + 0) * 256 + 16 * nt + l15];
        bb.y = fatt[base + (size_t)(k4 + 2 * half + 1) * 256 + 16 * nt + l15];
        acc = wmma4(a, bb, acc);
    }
#pragma unroll
    for (int r = 0; r < 8; ++r) {
        const int m = 16 * mt + r + 8 * half;
        y[base + (size_t)m * 256 + 16 * nt + l15] = acc[r];
    }
}

// ---------------------------------------------------------------------------
// Kernel 5: z[b,o,l] = sum_c Wo[o,c] * y[b,c,l]   (M=512,N=256,K=256)
// No bias here: bo is added in the spatial domain (doesn't commute with P^T).
// 4096 waves = 512 blocks x 8 waves.
// ---------------------------------------------------------------------------
__global__ void proj_out_kernel(const float* __restrict__ y,
                                const float* __restrict__ Wo,
                                float* __restrict__ z) {
    const int lane = threadIdx.x & 31;
    const int wave = threadIdx.x >> 5;
    const int half = lane >> 4;
    const int l15  = lane & 15;

    const int id = blockIdx.x * 8 + wave;        // 0..4095
    const int b  = id >> 9;
    const int mt = (id >> 4) & 31;
    const int nt = id & 15;

    const size_t ybase = (size_t)b * 256 * 256;
    v8f acc = {};
    for (int c4 = 0; c4 < 256; c4 += 4) {
        v2f a = *(const v2f*)(Wo + (size_t)(16 * mt + l15) * 256 + c4 + 2 * half);
        v2f bb;
        bb.x = y[ybase + (size_t)(c4 + 2 * half + 0) * 256 + 16 * nt + l15];
        bb.y = y[ybase + (size_t)(c4 + 2 * half + 1) * 256 + 16 * nt + l15];
        acc = wmma4(a, bb, acc);
    }
#pragma unroll
    for (int r = 0; r < 8; ++r) {
        const int o = 16 * mt + r + 8 * half;
        z[((size_t)b * 512 + o) * 256 + 16 * nt + l15] = acc[r];
    }
}

// ---------------------------------------------------------------------------
// Kernel 6: inverse separable DCT + residual + bo.
// One block per (b,o) image.  U(16x128) = Z @ Dv^T (wave w does cols 16w..),
// then Out = Dh @ U + x + bo  (wave w does row strip 16w..16w+16).
// x residual read and out write are 256 MB streams -> non-temporal.
// ---------------------------------------------------------------------------
__global__ void inv_dct_kernel(const float* __restrict__ z,
                               const float* __restrict__ x,
                               const float* __restrict__ Dh,
                               const float* __restrict__ Dv,
                               const float* __restrict__ bo,
                               float* __restrict__ out) {
    __shared__ float Ulds[16 * 128];

    const int lane = threadIdx.x & 31;
    const int wave = threadIdx.x >> 5;
    const int half = lane >> 4;
    const int l15  = lane & 15;

    const int boi = blockIdx.x;                  // b*512 + o
    const int o   = boi & 511;
    const float* __restrict__ Z = z + (size_t)boi * 256;

    // ---- U tile: cols [16w, 16w+16) ----
    v8f acc = {};
#pragma unroll
    for (int k4 = 0; k4 < 16; k4 += 4) {
        v2f a  = *(const v2f*)(Z + l15 * 16 + k4 + 2 * half);                // A[u][v]=Z[u,v]
        v2f bb = *(const v2f*)(Dv + (16 * wave + l15) * 16 + k4 + 2 * half); // B[v][w']=Dv[w',v]
        acc = wmma4(a, bb, acc);
    }
#pragma unroll
    for (int r = 0; r < 8; ++r)
        Ulds[(r + 8 * half) * 128 + 16 * wave + l15] = acc[r];
    __syncthreads();

    // ---- Out strip: rows [16w, 16w+16), all 8 column tiles ----
    const float bias = bo[o];
    const float* __restrict__ Xim = x + (size_t)boi * 16384;
    float* __restrict__ Oim = out + (size_t)boi * 16384;

    for (int j = 0; j < 8; ++j) {
        v8f acc2 = {};
#pragma unroll
        for (int k4 = 0; k4 < 16; k4 += 4) {
            v2f a = *(const v2f*)(Dh + (16 * wave + l15) * 16 + k4 + 2 * half); // A[h][u]=Dh[h,u]
            v2f bb;                                                            // B[u][w']=U[u,w']
            bb.x = Ulds[(k4 + 2 * half + 0) * 128 + 16 * j + l15];
            bb.y = Ulds[(k4 + 2 * half + 1) * 128 + 16 * j + l15];
            acc2 = wmma4(a, bb, acc2);
        }
#pragma unroll
        for (int r = 0; r < 8; ++r) {
            const int idx = (16 * wave + r + 8 * half) * 128 + 16 * j + l15;
            nt_store_f(Oim + idx, acc2[r] + nt_load_f(Xim + idx) + bias);
        }
    }
}

// ---------------------------------------------------------------------------
// Launch
// ---------------------------------------------------------------------------
extern "C" void kernel_launch(void* const* d_in, const int* in_sizes, int n_in,
                              void* d_out, int out_size, void* d_ws, size_t ws_size,
                              hipStream_t stream) {
    const float* x  = (const float*)d_in[0];
    const float* Wq = (const float*)d_in[1];
    const float* bq = (const float*)d_in[2];
    const float* Wk = (const float*)d_in[3];
    const float* bk = (const float*)d_in[4];
    const float* Wv = (const float*)d_in[5];
    const float* bv = (const float*)d_in[6];
    const float* Wo = (const float*)d_in[7];
    const float* bo = (const float*)d_in[8];
    float* out = (float*)d_out;

    // workspace layout (floats)
    float* w    = (float*)d_ws;
    float* Dh   = w;                 // 128*16
    float* Dv   = Dh + 2048;         // 128*16
    float* xf   = Dv + 2048;         // 8*512*256 = 1048576
    float* q    = xf + 1048576;      // 8*256*256 = 524288
    float* kk   = q  + 524288;
    float* v    = kk + 524288;
    float* fatt = v  + 524288;
    float* y    = fatt + 524288;
    float* z    = y  + 524288;       // 8*512*256 = 1048576

    build_basis_kernel<<<1, 256, 0, stream>>>(Dh, Dv);
    fwd_dct_kernel<<<4096, 256, 0, stream>>>(x, Dh, Dv, xf);
    proj_qkv_kernel<<<256, 256, 0, stream>>>(xf, Wq, bq, Wk, bk, Wv, bv, q, kk, v);
    fatt_kernel<<<256, 256, 0, stream>>>(q, kk, fatt);
    attn_apply_kernel<<<256, 256, 0, stream>>>(v, fatt, y);
    proj_out_kernel<<<512, 256, 0, stream>>>(y, Wo, z);
    inv_dct_kernel<<<4096, 256, 0, stream>>>(z, x, Dh, Dv, bo, out);
}